// MyClass_69114613730060
// MI455X (gfx1250) — compile-verified
//
#include <hip/hip_runtime.h>
#include <math.h>

// ---------------------------------------------------------------------------
// CARAFE upsample, fully fused for MI455X (gfx1250, wave32).
//
// Rank-1 fusion: comp = comp_w[c]*feat + comp_b[c]  =>  the 64-channel 3x3
// conv collapses to a small per-pixel GEMM against precomputed effective
// weights. The bias (enc_b) is folded in as a K column with A=1, so the
// WMMA accumulator starts at inline zero (no broadcast movs).
//
// Two GEMM shapes, selected by a *uniform* branch per block:
//   interior (91% of blocks): K=12  -> 3 x v_wmma_f32_16x16x4_f32 per N-tile
//       cols: 9 feat taps | enc_b + sum(B_tap) | 0 | 0      (no border VALU)
//   border ring:             K=20  -> 5 WMMAs per N-tile
//       cols: 9 feat taps | enc_b | 9 in-bounds indicators | 0
// 7 N-tiles cover the 100 (padded 112) mask channels.
//
// Epilogue: lane -> (pixel, subrow): 25 ds_load_b64 gathers (both subcols),
// dual softmax over 25, 5x5 reassembly, one coalesced b64 store per lane.
// Kernel is HBM-bound (~42 MB traffic: read x 8.4 MB + write out 33.5 MB).
// ---------------------------------------------------------------------------

typedef float v2f __attribute__((ext_vector_type(2)));
typedef float v8f __attribute__((ext_vector_type(8)));

#define FH   512      // feat height/width
#define XW   513      // x row stride
#define NCH  100      // mask channels
#define NPAD 112      // padded to 7*16
#define NT   7        // N tiles of 16
#define KT_I 3        // interior K steps (K=12)
#define KT_B 5        // border K steps (K=20)

// ws float layout: interior B-frags [0, 1344), border B-frags [1344, 3584)
#define WS_INT 0
#define WS_BRD (KT_I * NT * 64)
#define WS_TOTAL (WS_BRD + KT_B * NT * 64)

// ---------------------------------------------------------------------------
__device__ __forceinline__ float enc_dot(const float* __restrict__ enc_w,
                                         const float* __restrict__ v,
                                         int o, int tap) {
  float acc = 0.0f;
  for (int c = 0; c < 64; ++c)
    acc += enc_w[(o * 64 + c) * 9 + tap] * v[c];
  return acc;
}

// Fold enc_w x {comp_w, comp_b} into effective weights stored directly in the
// WMMA B-fragment lane layout: VGPR v, lane l: K = 4*kt + v + 2*(l>=16),
// N = nt*16 + (l&15).
__global__ void carafe_setup(const float* __restrict__ comp_w,
                             const float* __restrict__ comp_b,
                             const float* __restrict__ enc_w,
                             const float* __restrict__ enc_b,
                             float* __restrict__ ws) {
  const int t = threadIdx.x;
  // ---- interior table: K=12 ----
  for (int e = t; e < KT_I * NT * 64; e += blockDim.x) {
    const int l  = e & 31;
    const int v  = (e >> 5) & 1;
    const int nt = (e >> 6) % NT;
    const int kt = e / (NT * 64);
    const int K  = 4 * kt + v + 2 * (l >> 4);
    const int N  = nt * 16 + (l & 15);
    float val = 0.0f;
    if (N < NCH) {
      if (K < 9) {
        val = enc_dot(enc_w, comp_w, N, K);            // W_eff tap
      } else if (K == 9) {
        float acc = enc_b[N];                          // full bias: all taps in
        for (int tap = 0; tap < 9; ++tap)
          acc += enc_dot(enc_w, comp_b, N, tap);
        val = acc;
      }
    }
    ws[WS_INT + e] = val;
  }
  // ---- border table: K=20 ----
  for (int e = t; e < KT_B * NT * 64; e += blockDim.x) {
    const int l  = e & 31;
    const int v  = (e >> 5) & 1;
    const int nt = (e >> 6) % NT;
    const int kt = e / (NT * 64);
    const int K  = 4 * kt + v + 2 * (l >> 4);
    const int N  = nt * 16 + (l & 15);
    float val = 0.0f;
    if (N < NCH) {
      if (K < 9)        val = enc_dot(enc_w, comp_w, N, K);      // W_eff tap
      else if (K == 9)  val = enc_b[N];                          // bias column
      else if (K < 19)  val = enc_dot(enc_w, comp_b, N, K - 10); // indicator tap
    }
    ws[WS_BRD + e] = val;
  }
}

// ---------------------------------------------------------------------------
template <int KTN>
__device__ __forceinline__ void mask_gemm(const float* __restrict__ Btab,
                                          const float* aval, int lane,
                                          float* __restrict__ smrow) {
  const int mrow = (lane >> 4) * 8;    // C/D layout: VGPR r -> M = r + 8*(lane>=16)
  const int nlo  = lane & 15;
#pragma unroll
  for (int nt = 0; nt < NT; ++nt) {
    v8f c = {};                        // accumulator seeded with inline zero
#pragma unroll
    for (int kt = 0; kt < KTN; ++kt) {
      v2f a; a.x = aval[2 * kt]; a.y = aval[2 * kt + 1];
      v2f b;
      b.x = Btab[((kt * NT + nt) * 2 + 0) * 32 + lane];
      b.y = Btab[((kt * NT + nt) * 2 + 1) * 32 + lane];
      c = __builtin_amdgcn_wmma_f32_16x16x4_f32(false, a, false, b,
                                                (short)0, c, false, false);
    }
    const int ncol = nt * 16 + nlo;
#pragma unroll
    for (int r = 0; r < 8; ++r)
      smrow[(mrow + r) * NPAD + ncol] = c[r];
  }
}

// ---------------------------------------------------------------------------
// Block = 256 threads = 8 waves; covers 8(h) x 16(w) feat pixels ->
// 16 x 32 output pixels. Grid: (32 wtiles, 64 htiles, 8 batch).
// ---------------------------------------------------------------------------
__global__ void __launch_bounds__(256)
carafe_main(const float* __restrict__ x,
            const float* __restrict__ conv_w,
            const float* __restrict__ conv_b,
            const float* __restrict__ ws,
            float* __restrict__ out) {
  __shared__ float sfeat[12][20];          // feat rows h0-2..h0+9, cols w0-2..w0+17
  __shared__ float smask[8][16][NPAD];     // per-wave mask tiles (57344 B)

  const int tid  = threadIdx.x;
  const int lane = tid & 31;
  const int wid  = tid >> 5;

  const int w0 = blockIdx.x * 16;
  const int h0 = blockIdx.y * 8;
  const int n  = blockIdx.z;

  // uniform (scalar) interior test: tile rows/cols h0-2..h0+9 / w0-2..w0+17
  const bool interior = (blockIdx.x >= 1) & (blockIdx.x <= 30) &
                        (blockIdx.y >= 1) & (blockIdx.y <= 62);

  const float cw00 = conv_w[0], cw01 = conv_w[1];
  const float cw10 = conv_w[2], cw11 = conv_w[3];
  const float cb   = conv_b[0];
  const float* xb  = x + (size_t)n * XW * XW;

  // --- cooperative feat tile (leading valid 2x2 conv) ----------------------
  if (interior) {
    for (int e = tid; e < 12 * 20; e += 256) {
      const int r = e / 20, c = e % 20;
      const float* p = xb + (size_t)(h0 - 2 + r) * XW + (w0 - 2 + c);
      sfeat[r][c] = p[0] * cw00 + p[1] * cw01 + p[XW] * cw10 + p[XW + 1] * cw11 + cb;
    }
  } else {
    for (int e = tid; e < 12 * 20; e += 256) {
      const int r = e / 20, c = e % 20;
      const int fr = h0 - 2 + r, fc = w0 - 2 + c;
      float v = 0.0f;
      if (fr >= 0 && fr < FH && fc >= 0 && fc < FH) {
        const float* p = xb + (size_t)fr * XW + fc;
        v = p[0] * cw00 + p[1] * cw01 + p[XW] * cw10 + p[XW + 1] * cw11 + cb;
      }
      sfeat[r][c] = v;
    }
  }
  __syncthreads();

  // --- per-wave mask GEMM --------------------------------------------------
  const int M = lane & 15;                 // pixel within strip
  const int khalf = (lane >> 4) * 2;       // lanes 16-31 hold K+2, K+3
  float* smrow = &smask[wid][0][0];

  if (interior) {
    float aval[2 * KT_I];                  // K=12: taps | 1.0 | 0 | 0
#pragma unroll
    for (int j = 0; j < 2 * KT_I; ++j) {
      const int K = (j >> 1) * 4 + (j & 1) + khalf;
      float v = 0.0f;
      if (K < 9)        v = sfeat[wid + 2 + K / 3 - 1][M + 2 + K % 3 - 1];
      else if (K == 9)  v = 1.0f;
      aval[j] = v;
    }
    mask_gemm<KT_I>(ws + WS_INT, aval, lane, smrow);
  } else {
    const int h = h0 + wid, w = w0 + M;
    float aval[2 * KT_B];                  // K=20: taps | 1.0 | indicators | 0
#pragma unroll
    for (int j = 0; j < 2 * KT_B; ++j) {
      const int K = (j >> 1) * 4 + (j & 1) + khalf;
      float v = 0.0f;
      if (K < 9) {
        v = sfeat[wid + 2 + K / 3 - 1][M + 2 + K % 3 - 1];
      } else if (K == 9) {
        v = 1.0f;
      } else if (K < 19) {
        const int tap = K - 10;
        const int hy = h + tap / 3 - 1, wx = w + tap % 3 - 1;
        v = (hy >= 0 && hy < FH && wx >= 0 && wx < FH) ? 1.0f : 0.0f;
      }
      aval[j] = v;
    }
    mask_gemm<KT_B>(ws + WS_BRD, aval, lane, smrow);
  }
  __syncthreads();

  // --- dual softmax over 25 + 5x5 reassembly -------------------------------
  // lane -> (pixel pm, subrow sa); each lane covers both subcols via b64 ops.
  {
    const int pm = lane & 15;
    const int sa = lane >> 4;
    // channels 4*kk + 2*sa + {0,1} are adjacent -> float2 LDS gathers
    const v2f* mrow2 = (const v2f*)(&smask[wid][pm][0]);
    float m0[25], m1[25];
    float mx0 = -1e30f, mx1 = -1e30f;
#pragma unroll
    for (int kk = 0; kk < 25; ++kk) {
      v2f mv = mrow2[kk * 2 + sa];
      m0[kk] = mv.x; m1[kk] = mv.y;
      mx0 = fmaxf(mx0, mv.x); mx1 = fmaxf(mx1, mv.y);
    }
    float s0 = 0.0f, s1 = 0.0f;
#pragma unroll
    for (int kk = 0; kk < 25; ++kk) {
      m0[kk] = __expf(m0[kk] - mx0); s0 += m0[kk];
      m1[kk] = __expf(m1[kk] - mx1); s1 += m1[kk];
    }
    float acc0 = 0.0f, acc1 = 0.0f;
#pragma unroll
    for (int kk = 0; kk < 25; ++kk) {      // patches[k=i*5+j] = feat[h+i-2, w+j-2]
      const float f = sfeat[wid + kk / 5][pm + kk % 5];
      acc0 += m0[kk] * f;
      acc1 += m1[kk] * f;
    }
    v2f o2;
    o2.x = acc0 / s0;
    o2.y = acc1 / s1;
    const int oy = 2 * (h0 + wid) + sa;
    const int ox = 2 * (w0 + pm);          // even -> 8B aligned, coalesced b64
    *(v2f*)&out[((size_t)n * 1024 + oy) * 1024 + ox] = o2;
  }
}

// ---------------------------------------------------------------------------
extern "C" void kernel_launch(void* const* d_in, const int* in_sizes, int n_in,
                              void* d_out, int out_size, void* d_ws, size_t ws_size,
                              hipStream_t stream) {
  (void)in_sizes; (void)n_in; (void)out_size; (void)ws_size;
  const float* x      = (const float*)d_in[0];
  const float* conv_w = (const float*)d_in[1];
  const float* conv_b = (const float*)d_in[2];
  const float* comp_w = (const float*)d_in[3];
  const float* comp_b = (const float*)d_in[4];
  const float* enc_w  = (const float*)d_in[5];
  const float* enc_b  = (const float*)d_in[6];
  float* out = (float*)d_out;
  float* ws  = (float*)d_ws;

  carafe_setup<<<1, 256, 0, stream>>>(comp_w, comp_b, enc_w, enc_b, ws);

  dim3 grid(32, 64, 8);   // w-tiles, h-tiles(8 rows), batch
  carafe_main<<<grid, 256, 0, stream>>>(x, conv_w, conv_b, ws, out);
}